// Decoder_21912923144797
// MI455X (gfx1250) — compile-verified
//
#include <hip/hip_runtime.h>
#include <cmath>

typedef __attribute__((ext_vector_type(2))) float v2f;
typedef __attribute__((ext_vector_type(8))) float v8f;

constexpr int kH = 1024;
constexpr int kE = 512;
constexpr int kV = 50257;
constexpr int kB = 64;
constexpr int kS = 1024;
constexpr int kD = 2 * kH;       // 2048 encoder feature dim
constexpr int kX = 2 * kH + kE;  // 2560 x_in dim
constexpr int kG = 4 * kH;       // 4096 gates
constexpr int CHUNK = 128;       // seq rows per attention block
constexpr int NCH = kS / CHUNK;  // 8 chunks per batch

// ---------------------------------------------------------------- kernel 1
// hdot[b] = dot(hidden[b,:], We[2H:3H]) + be  (the hidden part of the energy
// is identical for every s, so hoist it out of the 536MB stream).
__global__ void k_hdot(const float* __restrict__ hidden,
                       const float* __restrict__ We,
                       const float* __restrict__ be,
                       float* __restrict__ hdot) {
    __shared__ float red[256];
    const int b = blockIdx.x, tid = threadIdx.x;
    const float* h = hidden + (size_t)b * kH;
    const float* w = We + kD;
    float s = 0.f;
    for (int j = tid; j < kH; j += 256) s += h[j] * w[j];
    red[tid] = s;
    __syncthreads();
    for (int off = 128; off >= 1; off >>= 1) {
        if (tid < off) red[tid] += red[tid + off];
        __syncthreads();
    }
    if (tid == 0) hdot[b] = red[0] + be[0];
}

// ---------------------------------------------------------------- kernel 2
// One block per (batch, 128-row chunk). Single HBM pass over encoder_states:
// phase1 energies (relu), phase2 chunk max + expsum, phase3 weighted partial
// context (chunk re-read hits L2). Emits (m, l, ctx_partial[2048]).
__global__ void k_attn_partial(const float* __restrict__ es,
                               const float* __restrict__ We,
                               const float* __restrict__ hdot,
                               float* __restrict__ pm,
                               float* __restrict__ pl,
                               float* __restrict__ pctx) {
    __shared__ float sWe[kD];
    __shared__ float sE[CHUNK];
    __shared__ float red[256];
    const int b = blockIdx.x / NCH, ch = blockIdx.x % NCH;
    const int tid = threadIdx.x, lane = tid & 31, wid = tid >> 5;

    for (int j = tid; j < kD; j += 256) sWe[j] = We[j];
    __syncthreads();

    const float hd = hdot[b];
    const float* base = es + ((size_t)b * kS + (size_t)ch * CHUNK) * kD;

    // phase 1: one wave computes 16 row-energies (strided dot + shuffle reduce)
    for (int i = 0; i < CHUNK / 8; ++i) {
        const int r = wid * (CHUNK / 8) + i;
        const float* row = base + (size_t)r * kD;
        float s = 0.f;
        for (int jj = 0; jj < kD / 32; ++jj) {
            const int idx = lane + 32 * jj;
            s += row[idx] * sWe[idx];
        }
        for (int off = 16; off >= 1; off >>= 1) s += __shfl_xor(s, off, 32);
        if (lane == 0) sE[r] = fmaxf(s + hd, 0.f);
    }
    __syncthreads();

    // phase 2: chunk max and exp-sum
    red[tid] = (tid < CHUNK) ? sE[tid] : -INFINITY;
    __syncthreads();
    for (int off = 128; off >= 1; off >>= 1) {
        if (tid < off) red[tid] = fmaxf(red[tid], red[tid + off]);
        __syncthreads();
    }
    const float m = red[0];
    __syncthreads();
    const float w = (tid < CHUNK) ? __expf(sE[tid] - m) : 0.f;
    red[tid] = w;
    __syncthreads();
    for (int off = 128; off >= 1; off >>= 1) {
        if (tid < off) red[tid] += red[tid + off];
        __syncthreads();
    }
    const float l = red[0];
    if (tid < CHUNK) sE[tid] = w;  // overwrite energies with weights
    __syncthreads();

    const int part = b * NCH + ch;
    if (tid == 0) { pm[part] = m; pl[part] = l; }

    // phase 3: ctx_partial[d] = sum_s w[s] * es[s][d]; thread owns 8 contiguous d
    float a0 = 0, a1 = 0, a2 = 0, a3 = 0, a4 = 0, a5 = 0, a6 = 0, a7 = 0;
    for (int s = 0; s < CHUNK; ++s) {
        const float ws = sE[s];
        const float4* p = ((const float4*)(base + (size_t)s * kD)) + tid * 2;
        const float4 r0 = p[0], r1 = p[1];
        a0 += ws * r0.x; a1 += ws * r0.y; a2 += ws * r0.z; a3 += ws * r0.w;
        a4 += ws * r1.x; a5 += ws * r1.y; a6 += ws * r1.z; a7 += ws * r1.w;
    }
    float4* outp = ((float4*)(pctx + (size_t)part * kD)) + tid * 2;
    outp[0] = make_float4(a0, a1, a2, a3);
    outp[1] = make_float4(a4, a5, a6, a7);
}

// ---------------------------------------------------------------- kernel 3
// Combine per-chunk partials into the softmax-normalized context, and gather
// the token embedding: x_in[b] = [ctx(2048) | emb[x_tok[b]](512)].
__global__ void k_attn_reduce(const float* __restrict__ pm,
                              const float* __restrict__ pl,
                              const float* __restrict__ pctx,
                              const int* __restrict__ x_tok,
                              const float* __restrict__ emb,
                              float* __restrict__ x_in) {
    __shared__ float sM[NCH], sL[NCH];
    const int b = blockIdx.x, tid = threadIdx.x;
    if (tid < NCH) { sM[tid] = pm[b * NCH + tid]; sL[tid] = pl[b * NCH + tid]; }
    __syncthreads();
    float M = -INFINITY;
    for (int i = 0; i < NCH; ++i) M = fmaxf(M, sM[i]);
    float L = 0.f;
    for (int i = 0; i < NCH; ++i) L += sL[i] * __expf(sM[i] - M);
    const float invL = 1.f / L;
    float sc[NCH];
    for (int i = 0; i < NCH; ++i) sc[i] = __expf(sM[i] - M) * invL;

    for (int d = tid; d < kD; d += 256) {
        float s = 0.f;
        for (int i = 0; i < NCH; ++i)
            s += sc[i] * pctx[(size_t)(b * NCH + i) * kD + d];
        x_in[(size_t)b * kX + d] = s;
    }
    const int tok = x_tok[b];
    for (int j = tid; j < kE; j += 256)
        x_in[(size_t)b * kX + kD + j] = emb[(size_t)tok * kE + j];
}

// ---------------------------------------------------------------- kernel 4
// gates[64,4096] = x_in @ W_ih^T + hidden @ W_hh^T + b_ih + b_hh via
// V_WMMA_F32_16X16X4_F32. One wave per 16-wide gate tile; M=64 -> 4 acc tiles.
// A frag (16x4): lane%16 = M row, lane/16 selects K pair. B frag (4x16):
// lane%16 = N col -> loading W[n][k..k+1] yields B[K][N] = W^T as required.
__global__ void k_gates(const float* __restrict__ x_in,
                        const float* __restrict__ hidden,
                        const float* __restrict__ W_ih,
                        const float* __restrict__ W_hh,
                        const float* __restrict__ b_ih,
                        const float* __restrict__ b_hh,
                        float* __restrict__ gates) {
    const int lane = threadIdx.x;
    const int n0 = blockIdx.x * 16;
    const int nrow = n0 + (lane & 15);
    const int mrow = lane & 15;
    const int khalf = (lane >> 4) * 2;

    v8f acc0 = {}, acc1 = {}, acc2 = {}, acc3 = {};

    const float* wb = W_ih + (size_t)nrow * kX + khalf;
    const float* a0p = x_in + (size_t)(0 * 16 + mrow) * kX + khalf;
    const float* a1p = x_in + (size_t)(1 * 16 + mrow) * kX + khalf;
    const float* a2p = x_in + (size_t)(2 * 16 + mrow) * kX + khalf;
    const float* a3p = x_in + (size_t)(3 * 16 + mrow) * kX + khalf;
    for (int k = 0; k < kX; k += 4) {
        const v2f bfrag = *(const v2f*)(wb + k);
        const v2f a0 = *(const v2f*)(a0p + k);
        const v2f a1 = *(const v2f*)(a1p + k);
        const v2f a2 = *(const v2f*)(a2p + k);
        const v2f a3 = *(const v2f*)(a3p + k);
        acc0 = __builtin_amdgcn_wmma_f32_16x16x4_f32(false, a0, false, bfrag, (short)0, acc0, false, false);
        acc1 = __builtin_amdgcn_wmma_f32_16x16x4_f32(false, a1, false, bfrag, (short)0, acc1, false, false);
        acc2 = __builtin_amdgcn_wmma_f32_16x16x4_f32(false, a2, false, bfrag, (short)0, acc2, false, false);
        acc3 = __builtin_amdgcn_wmma_f32_16x16x4_f32(false, a3, false, bfrag, (short)0, acc3, false, false);
    }

    const float* wh = W_hh + (size_t)nrow * kH + khalf;
    const float* h0p = hidden + (size_t)(0 * 16 + mrow) * kH + khalf;
    const float* h1p = hidden + (size_t)(1 * 16 + mrow) * kH + khalf;
    const float* h2p = hidden + (size_t)(2 * 16 + mrow) * kH + khalf;
    const float* h3p = hidden + (size_t)(3 * 16 + mrow) * kH + khalf;
    for (int k = 0; k < kH; k += 4) {
        const v2f bfrag = *(const v2f*)(wh + k);
        const v2f a0 = *(const v2f*)(h0p + k);
        const v2f a1 = *(const v2f*)(h1p + k);
        const v2f a2 = *(const v2f*)(h2p + k);
        const v2f a3 = *(const v2f*)(h3p + k);
        acc0 = __builtin_amdgcn_wmma_f32_16x16x4_f32(false, a0, false, bfrag, (short)0, acc0, false, false);
        acc1 = __builtin_amdgcn_wmma_f32_16x16x4_f32(false, a1, false, bfrag, (short)0, acc1, false, false);
        acc2 = __builtin_amdgcn_wmma_f32_16x16x4_f32(false, a2, false, bfrag, (short)0, acc2, false, false);
        acc3 = __builtin_amdgcn_wmma_f32_16x16x4_f32(false, a3, false, bfrag, (short)0, acc3, false, false);
    }

    const float bias = b_ih[nrow] + b_hh[nrow];
    const int mbase = 8 * (lane >> 4);  // C/D layout: VGPR v -> M = v + 8*(lane/16)
    for (int v = 0; v < 8; ++v) {
        gates[(size_t)(0 * 16 + mbase + v) * kG + nrow] = acc0[v] + bias;
        gates[(size_t)(1 * 16 + mbase + v) * kG + nrow] = acc1[v] + bias;
        gates[(size_t)(2 * 16 + mbase + v) * kG + nrow] = acc2[v] + bias;
        gates[(size_t)(3 * 16 + mbase + v) * kG + nrow] = acc3[v] + bias;
    }
}

// ---------------------------------------------------------------- kernel 5
// LSTM elementwise step (PyTorch gate order i,f,g,o). Writes h_next / c_next
// directly into their d_out slots.
__global__ void k_lstm(const float* __restrict__ gates,
                       const float* __restrict__ cell,
                       float* __restrict__ out) {
    const int idx = blockIdx.x * 256 + threadIdx.x;  // [0, 64*1024)
    const int b = idx >> 10, h = idx & (kH - 1);
    const float* g = gates + (size_t)b * kG;
    const float gi = g[h];
    const float gf = g[h + kH];
    const float gg = g[h + 2 * kH];
    const float go = g[h + 3 * kH];
    const float si = 1.f / (1.f + __expf(-gi));
    const float sf = 1.f / (1.f + __expf(-gf));
    const float so = 1.f / (1.f + __expf(-go));
    const float c = sf * cell[idx] + si * tanhf(gg);
    const float hn = so * tanhf(c);
    out[(size_t)kB * kV + idx] = hn;                    // h_next
    out[(size_t)kB * kV + (size_t)kB * kH + idx] = c;   // c_next
}

// ---------------------------------------------------------------- kernel 6
// predictions[64,V] = h_next @ Wf^T + bf via f32 WMMA. One wave per 16-wide
// vocab tile, K=1024 streamed; Wf (206MB) is the HBM bottleneck. Tail columns
// (V=50257) are clamped for loads, masked for stores.
__global__ void k_proj(const float* __restrict__ hnext,
                       const float* __restrict__ Wf,
                       const float* __restrict__ bfv,
                       float* __restrict__ pred) {
    const int lane = threadIdx.x;
    const int n0 = blockIdx.x * 16;
    const int n = n0 + (lane & 15);
    const int nrow = (n < kV) ? n : (kV - 1);
    const int mrow = lane & 15;
    const int khalf = (lane >> 4) * 2;

    v8f acc0 = {}, acc1 = {}, acc2 = {}, acc3 = {};

    const float* wb = Wf + (size_t)nrow * kH + khalf;
    const float* h0p = hnext + (size_t)(0 * 16 + mrow) * kH + khalf;
    const float* h1p = hnext + (size_t)(1 * 16 + mrow) * kH + khalf;
    const float* h2p = hnext + (size_t)(2 * 16 + mrow) * kH + khalf;
    const float* h3p = hnext + (size_t)(3 * 16 + mrow) * kH + khalf;
    for (int k = 0; k < kH; k += 4) {
        const v2f bfrag = *(const v2f*)(wb + k);
        const v2f a0 = *(const v2f*)(h0p + k);
        const v2f a1 = *(const v2f*)(h1p + k);
        const v2f a2 = *(const v2f*)(h2p + k);
        const v2f a3 = *(const v2f*)(h3p + k);
        acc0 = __builtin_amdgcn_wmma_f32_16x16x4_f32(false, a0, false, bfrag, (short)0, acc0, false, false);
        acc1 = __builtin_amdgcn_wmma_f32_16x16x4_f32(false, a1, false, bfrag, (short)0, acc1, false, false);
        acc2 = __builtin_amdgcn_wmma_f32_16x16x4_f32(false, a2, false, bfrag, (short)0, acc2, false, false);
        acc3 = __builtin_amdgcn_wmma_f32_16x16x4_f32(false, a3, false, bfrag, (short)0, acc3, false, false);
    }

    if (n < kV) {
        const float bias = bfv[n];
        const int mbase = 8 * (lane >> 4);
        for (int v = 0; v < 8; ++v) {
            pred[(size_t)(0 * 16 + mbase + v) * kV + n] = acc0[v] + bias;
            pred[(size_t)(1 * 16 + mbase + v) * kV + n] = acc1[v] + bias;
            pred[(size_t)(2 * 16 + mbase + v) * kV + n] = acc2[v] + bias;
            pred[(size_t)(3 * 16 + mbase + v) * kV + n] = acc3[v] + bias;
        }
    }
}

// ---------------------------------------------------------------- launch
extern "C" void kernel_launch(void* const* d_in, const int* in_sizes, int n_in,
                              void* d_out, int out_size, void* d_ws, size_t ws_size,
                              hipStream_t stream) {
    const float* es     = (const float*)d_in[0];
    const int*   x_tok  = (const int*)d_in[1];
    const float* hidden = (const float*)d_in[2];
    const float* cell   = (const float*)d_in[3];
    const float* emb    = (const float*)d_in[4];
    const float* We     = (const float*)d_in[5];
    const float* be     = (const float*)d_in[6];
    const float* W_ih   = (const float*)d_in[7];
    const float* W_hh   = (const float*)d_in[8];
    const float* b_ih   = (const float*)d_in[9];
    const float* b_hh   = (const float*)d_in[10];
    const float* Wf     = (const float*)d_in[11];
    const float* bf     = (const float*)d_in[12];
    float* out = (float*)d_out;

    // workspace layout (floats): ~5.9 MB total
    float* ws    = (float*)d_ws;
    float* hdot  = ws;                          // 64
    float* x_in  = hdot + 64;                   // 64*2560
    float* gates = x_in + (size_t)kB * kX;      // 64*4096
    float* pm    = gates + (size_t)kB * kG;     // 64*8
    float* pl    = pm + kB * NCH;               // 64*8
    float* pctx  = pl + kB * NCH;               // 64*8*2048

    k_hdot        <<<kB,            256, 0, stream>>>(hidden, We, be, hdot);
    k_attn_partial<<<kB * NCH,      256, 0, stream>>>(es, We, hdot, pm, pl, pctx);
    k_attn_reduce <<<kB,            256, 0, stream>>>(pm, pl, pctx, x_tok, emb, x_in);
    k_gates       <<<kG / 16,        32, 0, stream>>>(x_in, hidden, W_ih, W_hh, b_ih, b_hh, gates);
    k_lstm        <<<(kB * kH)/256, 256, 0, stream>>>(gates, cell, out);
    k_proj        <<<(kV + 15)/16,   32, 0, stream>>>(out + (size_t)kB * kV, Wf, bf, out);
}